// NGCFLayer_our1_52561809769216
// MI455X (gfx1250) — compile-verified
//
#include <hip/hip_runtime.h>

// ---------------------------------------------------------------------------
// NGCF layer for MI455X (gfx1250, wave32).
//   h_u = W1 f_u ; h_i = W1 f_i                          (self loop)
//   h_i += segsum( norm_ui * (W1 f_u[s] + W2 (f_u[s]*f_i[d]) + b2), d )
//   h_u += segsum( norm_iu * (W1 f_i[s] + W2 (f_i[s]*f_u[d]) + b2), d )
//   out = l2norm(leakyrelu(h))
// All GEMMs via V_WMMA_F32_16X16X4_F32 (exact f32). Memory-bound workload:
// ~2.3 GB of mostly-random traffic vs ~19 GFLOP.
// ---------------------------------------------------------------------------

typedef float v2f __attribute__((ext_vector_type(2)));
typedef float v8f __attribute__((ext_vector_type(8)));

#define LSTRIDE 68          // padded LDS row stride (floats): 68 % 64 = 4 -> no bank conflicts
#define WAVES_PER_BLOCK 8
#define BLOCK 256

__device__ __forceinline__ v8f wmma_f32(v2f a, v2f b, v8f c) {
  // D(16x16) = A(16x4) * B(4x16) + C, full f32.
  return __builtin_amdgcn_wmma_f32_16x16x4_f32(
      /*neg_a=*/false, a, /*neg_b=*/false, b,
      /*c_mod=*/(short)0, c, /*reuse_a=*/false, /*reuse_b=*/false);
}

// Stage a 64x64 row-major matrix into LDS with padded stride.
__device__ __forceinline__ void stage_matrix64(const float* __restrict__ g,
                                               float* __restrict__ lds, int tid) {
  for (int i = tid; i < 64 * 16; i += BLOCK) {
    const int r = i >> 4, c4 = (i & 15) << 2;
    const float4 w = *(const float4*)(g + r * 64 + c4);
    float* d = lds + r * LSTRIDE + c4;
    d[0] = w.x; d[1] = w.y; d[2] = w.z; d[3] = w.w;
  }
}

// ---------------------------------------------------------------------------
// Kernel 1: w1_out = feat @ W1^T + b1 ; h_out = w1_out (self-loop init).
// One wave computes a 16x64 output tile. nrows must be a multiple of 16
// (NU = NI = 150000 = 9375*16).
// ---------------------------------------------------------------------------
__global__ __launch_bounds__(BLOCK) void ngcf_node_linear(
    const float* __restrict__ feat, const float* __restrict__ W1,
    const float* __restrict__ b1, float* __restrict__ w1_out,
    float* __restrict__ h_out, int nrows) {
  __shared__ float Wlds[64 * LSTRIDE];
  __shared__ float blds[64];
  __shared__ float Ftile[WAVES_PER_BLOCK][16 * LSTRIDE];

  const int tid = threadIdx.x;
  const int lane = tid & 31;
  const int wave = tid >> 5;
  const int lo = lane & 15;   // col (B/C) or row (A) within tile
  const int hi = lane >> 4;   // half-wave select

  stage_matrix64(W1, Wlds, tid);               // B[k][n] = W1[n][k] -> store W1 as-is
  if (tid < 64) blds[tid] = b1[tid];

  const int tile = blockIdx.x * WAVES_PER_BLOCK + wave;
  const int ntiles = nrows >> 4;
  const bool active = tile < ntiles;           // wave-uniform (EXEC stays all-1s)
  const int row0 = tile << 4;

  if (active) {
    float* ft = &Ftile[wave][0];
#pragma unroll
    for (int i = 0; i < 8; ++i) {
      const int s = i * 32 + lane, m = s >> 4, c4 = (s & 15) << 2;
      const float4 f = *(const float4*)(feat + (size_t)(row0 + m) * 64 + c4);
      float* d = ft + m * LSTRIDE + c4;
      d[0] = f.x; d[1] = f.y; d[2] = f.z; d[3] = f.w;
    }
  }
  __syncthreads();

  if (active) {
    const float* ft = &Ftile[wave][0];
    v8f acc[4] = {};
#pragma unroll
    for (int k = 0; k < 16; ++k) {
      const int kb = (k << 2) + (hi << 1);                  // K base for this half-wave
      const v2f a = *(const v2f*)(ft + lo * LSTRIDE + kb);  // A: lane=M, VGPRs=K,K+1
#pragma unroll
      for (int nt = 0; nt < 4; ++nt) {
        const v2f b = *(const v2f*)(&Wlds[(nt * 16 + lo) * LSTRIDE + kb]);  // B: lane=N
        acc[nt] = wmma_f32(a, b, acc[nt]);
      }
    }
#pragma unroll
    for (int j = 0; j < 8; ++j) {
      const int m = j + (hi << 3);           // C/D: VGPR j holds row j + 8*hi
      const size_t rb = (size_t)(row0 + m) * 64;
#pragma unroll
      for (int nt = 0; nt < 4; ++nt) {
        const int col = nt * 16 + lo;
        const float v = acc[nt][j] + blds[col];
        w1_out[rb + col] = v;
        h_out[rb + col] = v;
      }
    }
  }
}

// ---------------------------------------------------------------------------
// Kernel 2: per edge e: msg = norm[e]*(w1_src[src] + (f_src[src]*f_dst[dst])@W2^T + b2)
//           h_acc[dst] += msg   (hardware f32 atomics == segment_sum)
// One wave per 16 edges. nedges must be a multiple of 16 (E = 1e6 = 62500*16).
// ---------------------------------------------------------------------------
__global__ __launch_bounds__(BLOCK) void ngcf_edge(
    const float* __restrict__ feat_src, const float* __restrict__ feat_dst,
    const float* __restrict__ w1_src, const float* __restrict__ W2,
    const float* __restrict__ b2, const float* __restrict__ norm,
    const int* __restrict__ src, const int* __restrict__ dst,
    float* __restrict__ h_acc, int nedges) {
  __shared__ float Wlds[64 * LSTRIDE];
  __shared__ float blds[64];
  __shared__ float UV[WAVES_PER_BLOCK][16 * LSTRIDE];
  __shared__ float WS[WAVES_PER_BLOCK][16 * LSTRIDE];

  const int tid = threadIdx.x;
  const int lane = tid & 31;
  const int wave = tid >> 5;
  const int lo = lane & 15;
  const int hi = lane >> 4;

  stage_matrix64(W2, Wlds, tid);
  if (tid < 64) blds[tid] = b2[tid];

  const int tile = blockIdx.x * WAVES_PER_BLOCK + wave;
  const int ntiles = nedges >> 4;
  const bool active = tile < ntiles;           // wave-uniform
  const int e0 = tile << 4;

  if (active) {
    float* uv = &UV[wave][0];
    float* ws = &WS[wave][0];
#pragma unroll
    for (int i = 0; i < 8; ++i) {
      const int s = i * 32 + lane, m = s >> 4, c4 = (s & 15) << 2;
      const int es = src[e0 + m];              // 16 lanes share an edge -> L1 hit
      const int ed = dst[e0 + m];
      const float4 a = *(const float4*)(feat_src + (size_t)es * 64 + c4);
      const float4 b = *(const float4*)(feat_dst + (size_t)ed * 64 + c4);
      const float4 w = *(const float4*)(w1_src + (size_t)es * 64 + c4);
      float* du = uv + m * LSTRIDE + c4;
      du[0] = a.x * b.x; du[1] = a.y * b.y; du[2] = a.z * b.z; du[3] = a.w * b.w;
      float* dw = ws + m * LSTRIDE + c4;
      dw[0] = w.x; dw[1] = w.y; dw[2] = w.z; dw[3] = w.w;
    }
  }
  __syncthreads();

  if (active) {
    const float* uv = &UV[wave][0];
    v8f acc[4] = {};
#pragma unroll
    for (int k = 0; k < 16; ++k) {
      const int kb = (k << 2) + (hi << 1);
      const v2f a = *(const v2f*)(uv + lo * LSTRIDE + kb);
#pragma unroll
      for (int nt = 0; nt < 4; ++nt) {
        const v2f b = *(const v2f*)(&Wlds[(nt * 16 + lo) * LSTRIDE + kb]);
        acc[nt] = wmma_f32(a, b, acc[nt]);
      }
    }
    const float* ws = &WS[wave][0];
#pragma unroll
    for (int j = 0; j < 8; ++j) {
      const int m = j + (hi << 3);
      const float nr = norm[e0 + m];
      const size_t db = (size_t)dst[e0 + m] * 64;
#pragma unroll
      for (int nt = 0; nt < 4; ++nt) {
        const int col = nt * 16 + lo;
        const float v = (acc[nt][j] + ws[m * LSTRIDE + col] + blds[col]) * nr;
        unsafeAtomicAdd(&h_acc[db + col], v);  // -> global_atomic_add_f32
      }
    }
  }
}

// ---------------------------------------------------------------------------
// Kernel 3: out = l2normalize(leakyrelu(h, 0.2), dim=1). One wave per row,
// 2 floats per lane, shfl_xor reduction over the 32 lanes.
// ---------------------------------------------------------------------------
__global__ __launch_bounds__(BLOCK) void ngcf_finish(const float* __restrict__ h,
                                                     float* __restrict__ out, int n) {
  const int lane = threadIdx.x & 31;
  const int row = blockIdx.x * (BLOCK / 32) + (threadIdx.x >> 5);
  if (row >= n) return;
  const float2 hv = *(const float2*)(h + (size_t)row * 64 + lane * 2);
  const float x = hv.x > 0.f ? hv.x : 0.2f * hv.x;
  const float y = hv.y > 0.f ? hv.y : 0.2f * hv.y;
  float ss = x * x + y * y;
#pragma unroll
  for (int off = 16; off > 0; off >>= 1) ss += __shfl_xor(ss, off, 32);
  const float inv = 1.0f / fmaxf(sqrtf(ss), 1e-12f);
  float2 o;
  o.x = x * inv;
  o.y = y * inv;
  *(float2*)(out + (size_t)row * 64 + lane * 2) = o;
}

// ---------------------------------------------------------------------------
extern "C" void kernel_launch(void* const* d_in, const int* in_sizes, int n_in,
                              void* d_out, int out_size, void* d_ws, size_t ws_size,
                              hipStream_t stream) {
  const float* feat_user = (const float*)d_in[0];
  const float* feat_item = (const float*)d_in[1];
  const float* W1 = (const float*)d_in[2];
  const float* b1 = (const float*)d_in[3];
  const float* W2 = (const float*)d_in[4];
  const float* b2 = (const float*)d_in[5];
  const float* norm_ui = (const float*)d_in[6];
  const float* norm_iu = (const float*)d_in[7];
  const int* ui_src = (const int*)d_in[8];
  const int* ui_dst = (const int*)d_in[9];
  const int* iu_src = (const int*)d_in[10];
  const int* iu_dst = (const int*)d_in[11];

  const int NU = in_sizes[0] / 64;
  const int NI = in_sizes[1] / 64;
  const int E = in_sizes[8];

  // Workspace: w1_u | w1_i | h_u | h_i  (4 * N * 64 floats ~ 153.6 MB)
  float* w1_u = (float*)d_ws;
  float* w1_i = w1_u + (size_t)NU * 64;
  float* h_u = w1_i + (size_t)NI * 64;
  float* h_i = h_u + (size_t)NU * 64;

  auto cdiv = [](int a, int b) { return (a + b - 1) / b; };

  ngcf_node_linear<<<cdiv(NU / 16, WAVES_PER_BLOCK), BLOCK, 0, stream>>>(
      feat_user, W1, b1, w1_u, h_u, NU);
  ngcf_node_linear<<<cdiv(NI / 16, WAVES_PER_BLOCK), BLOCK, 0, stream>>>(
      feat_item, W1, b1, w1_i, h_i, NI);

  ngcf_edge<<<cdiv(E / 16, WAVES_PER_BLOCK), BLOCK, 0, stream>>>(
      feat_user, feat_item, w1_u, W2, b2, norm_ui, ui_src, ui_dst, h_i, E);
  ngcf_edge<<<cdiv(E / 16, WAVES_PER_BLOCK), BLOCK, 0, stream>>>(
      feat_item, feat_user, w1_i, W2, b2, norm_iu, iu_src, iu_dst, h_u, E);

  float* out = (float*)d_out;
  ngcf_finish<<<cdiv(NU, BLOCK / 32), BLOCK, 0, stream>>>(h_u, out, NU);
  ngcf_finish<<<cdiv(NI, BLOCK / 32), BLOCK, 0, stream>>>(h_i, out + (size_t)NU * 64, NI);
}